// GatherNL_78022375899654
// MI455X (gfx1250) — compile-verified
//
#include <hip/hip_runtime.h>

// Problem constants (match reference)
#define TT 16
#define CC 3
#define HH 512
#define WW 512
#define PSZ 7
#define PTT 1
#define NQQ 262144

typedef __attribute__((ext_vector_type(4))) float v4f;

static constexpr int ELEMS_PER_PATCH = PTT * CC * PSZ * PSZ;        // 147
static constexpr int TOTAL_ELEMS     = NQQ * ELEMS_PER_PATCH;       // 38,535,168
static constexpr int OUT_ELEMS       = TT * CC * HH * WW;           // 12,582,912
static constexpr int OUT_VEC4        = OUT_ELEMS / 4;               // 3,145,728

// Scatter-add. One thread per patch element (defined first so the disasm
// snippet shows this kernel's body, letting us verify the atomic lowering).
//  patches: [N, 1, C, 7, 7] contiguous -> element e maps to
//    n = e / 147, c = (e % 147) / 49, i = (e % 49) / 7, j = e % 7
//  output index: ((t0*C + c)*H + (h0+i))*W + (w0+j)
// Patch loads: coalesced + non-temporal (streamed 154 MB must not evict the
// 50 MB output from L2). Atomic adds: relaxed, agent scope -> hardware
// global_atomic_add_f32 without return, resolved in L2.
__global__ void scatter_add_kernel(const float* __restrict__ patches,
                                   const int* __restrict__ qinds,
                                   float* __restrict__ out) {
    int e = blockIdx.x * blockDim.x + threadIdx.x;
    if (e >= TOTAL_ELEMS) return;

    int n  = e / ELEMS_PER_PATCH;            // magic-mul division
    int r  = e - n * ELEMS_PER_PATCH;        // 0..146
    int c  = r / (PSZ * PSZ);                // 0..2
    int r2 = r - c * (PSZ * PSZ);            // 0..48
    int i  = r2 / PSZ;                       // 0..6
    int j  = r2 - i * PSZ;                   // 0..6

    // (t,h,w) corner for this patch; 147 consecutive threads share one n,
    // so these loads are wave-broadcasts served from cache.
    int t0 = qinds[3 * n + 0];
    int h0 = qinds[3 * n + 1];
    int w0 = qinds[3 * n + 2];

    float v = __builtin_nontemporal_load(patches + e);

    int oidx = ((t0 * CC + c) * HH + (h0 + i)) * WW + (w0 + j);
    __hip_atomic_fetch_add(out + oidx, v, __ATOMIC_RELAXED,
                           __HIP_MEMORY_SCOPE_AGENT);
}

// d_out = vid2fill (full deterministic rewrite every call).
// 128-bit vector copy; source read non-temporally (read exactly once),
// destination stored with default RT hint so it lands/stays in L2 for the
// atomic phase.
__global__ void copy_out_kernel(const float* __restrict__ src,
                                float* __restrict__ dst) {
    int i = blockIdx.x * blockDim.x + threadIdx.x;
    if (i < OUT_VEC4) {
        v4f v = __builtin_nontemporal_load(((const v4f*)src) + i);
        ((v4f*)dst)[i] = v;
    }
}

extern "C" void kernel_launch(void* const* d_in, const int* in_sizes, int n_in,
                              void* d_out, int out_size, void* d_ws, size_t ws_size,
                              hipStream_t stream) {
    const float* vid2fill = (const float*)d_in[0];   // [16,3,512,512] f32 (zeros)
    const float* patches  = (const float*)d_in[1];   // [262144,1,3,7,7] f32
    const int*   qinds    = (const int*)d_in[2];     // [262144,3] i32
    float*       out      = (float*)d_out;           // [16,3,512,512] f32

    // Phase 1: initialize output with vid2fill (overwrites poison, makes the
    // launch idempotent across graph replays).
    {
        const int threads = 256;
        const int blocks  = (OUT_VEC4 + threads - 1) / threads;   // 12288
        copy_out_kernel<<<blocks, threads, 0, stream>>>(vid2fill, out);
    }

    // Phase 2: atomic scatter-add of all patch elements (same stream ->
    // ordered after the copy).
    {
        const int threads = 256;
        const int blocks  = (TOTAL_ELEMS + threads - 1) / threads; // 150528
        scatter_add_kernel<<<blocks, threads, 0, stream>>>(patches, qinds, out);
    }
}